// loss_attack_41764261986608
// MI455X (gfx1250) — compile-verified
//
#include <hip/hip_runtime.h>
#include <hip/hip_bf16.h>
#include <stdint.h>

typedef __attribute__((ext_vector_type(2))) float        v2f;
typedef __attribute__((ext_vector_type(8))) float        v8f;
typedef __attribute__((ext_vector_type(4))) unsigned int v4u;
typedef __attribute__((ext_vector_type(8))) int          v8i;
typedef __attribute__((ext_vector_type(4))) int          v4i;

#define NANCH  4194304
#define KTOP   200
#define GBOX   32
#define IOU_T  0.3f
#define NBINS  32768      // scores in (0.5,1) -> bits>>8 gives 32768 monotonic bins
#define CAP    2048

// workspace layout (uint32 units)
#define WS_HIST 0
#define WS_SEL  (NBINS)          // [0] = threshold bin
#define WS_CNT  (NBINS + 2)      // candidate counter
#define WS_CB   (NBINS + 4)      // candidate score bits
#define WS_CI   (NBINS + 4 + CAP)// candidate indices

__global__ void zero_ws(unsigned* __restrict__ ws, int n) {
    int i = blockIdx.x * blockDim.x + threadIdx.x;
    if (i < n) ws[i] = 0u;
}

// Pass A: stream logits, histogram score bits (only klass!=0 i.e. l1>l0).
__global__ void score_hist(const float* __restrict__ pred, unsigned* __restrict__ ws) {
    unsigned* hist = ws + WS_HIST;
    int stride = gridDim.x * blockDim.x;
    for (int i = blockIdx.x * blockDim.x + threadIdx.x; i < NANCH; i += stride) {
        const float2 lg = *(const float2*)(pred + (size_t)i * 6 + 4);
        if (i + stride < NANCH)
            __builtin_prefetch(pred + (size_t)(i + stride) * 6 + 4, 0, 1); // global_prefetch_b8
        if (lg.y > lg.x) {
            float s = 1.0f / (1.0f + __expf(lg.x - lg.y));  // max softmax prob, in (0.5,1)
            unsigned bits = __float_as_uint(s);
            int bin = (int)(bits >> 8) - 0x3F0000;
            bin = bin < 0 ? 0 : (bin >= NBINS ? NBINS - 1 : bin);
            atomicAdd(&hist[bin], 1u);
        }
    }
}

// Pass B: find largest bin T with suffix-count >= KTOP.
__global__ void select_threshold(unsigned* __restrict__ ws) {
    __shared__ unsigned psum[1024];
    const unsigned* hist = ws + WS_HIST;
    int t = threadIdx.x;
    unsigned s = 0;
    for (int b = 0; b < 32; ++b) s += hist[t * 32 + b];
    psum[t] = s;
    __syncthreads();
    if (t == 0) {
        unsigned cum = 0; int T = 0;
        for (int c = 1023; c >= 0; --c) {
            if (cum + psum[c] >= KTOP) {
                for (int b = c * 32 + 31; b >= c * 32; --b) {
                    cum += hist[b];
                    if (cum >= KTOP) { T = b; break; }
                }
                break;
            }
            cum += psum[c];
        }
        ws[WS_SEL] = (unsigned)T;
    }
}

// Pass C: compact all candidates at/above the threshold bin.
__global__ void compact_cand(const float* __restrict__ pred, unsigned* __restrict__ ws) {
    const unsigned T = ws[WS_SEL];
    unsigned* cnt = ws + WS_CNT;
    unsigned* cb  = ws + WS_CB;
    unsigned* ci  = ws + WS_CI;
    int stride = gridDim.x * blockDim.x;
    for (int i = blockIdx.x * blockDim.x + threadIdx.x; i < NANCH; i += stride) {
        const float2 lg = *(const float2*)(pred + (size_t)i * 6 + 4);
        if (lg.y > lg.x) {
            float s = 1.0f / (1.0f + __expf(lg.x - lg.y));
            unsigned bits = __float_as_uint(s);
            int bin = (int)(bits >> 8) - 0x3F0000;
            bin = bin < 0 ? 0 : (bin >= NBINS ? NBINS - 1 : bin);
            if ((unsigned)bin >= T) {
                unsigned pos = atomicAdd(cnt, 1u);
                if (pos < CAP) { cb[pos] = bits; ci[pos] = (unsigned)i; }
            }
        }
    }
}

// Final: exact top-K ranking, decode, IoU vs 32 GT boxes, loss; WMMA f32 reduction.
__global__ __launch_bounds__(256)
void final_loss(const float* __restrict__ pred, const float* __restrict__ anch,
                const float* __restrict__ y, const unsigned* __restrict__ ws,
                float* __restrict__ out) {
    __shared__ float    yb[4 * GBOX];
    __shared__ unsigned scb[CAP];
    __shared__ unsigned sci[CAP];
    __shared__ float    partial[256];
    __shared__ float    colsum[2];

    const int tid = threadIdx.x;
    const unsigned Mraw = ws[WS_CNT];
    const int M = (int)(Mraw < CAP ? Mraw : CAP);

#if __has_builtin(__builtin_amdgcn_tensor_load_to_lds)
    // Stage GT boxes (128 f32) into LDS with the Tensor Data Mover.
    if (tid == 0) {
        unsigned long long ga = (unsigned long long)(uintptr_t)y;
        v4u g0;
        g0.x = 1u;                                   // count=1, user mode
        g0.y = (unsigned)(uintptr_t)(void*)yb;       // lds_addr
        g0.z = (unsigned)(ga & 0xFFFFFFFFull);       // global_addr[31:0]
        g0.w = 0x80000000u | (unsigned)((ga >> 32) & 0x01FFFFFFull); // type=2 | addr[56:32]
        v8i g1;
        g1[0] = 0x00020000;          // data_size = 4B
        g1[1] = (int)(128u << 16);   // tensor_dim0 = 128
        g1[2] = (int)(1u  << 16);    // tensor_dim1 = 1
        g1[3] = (int)(128u << 16);   // tile_dim0 = 128
        g1[4] = 1;                   // tile_dim1 = 1
        g1[5] = 128;                 // tensor_dim0_stride
        g1[6] = (int)(128u << 16);   // tensor_dim1_stride (low 16)
        g1[7] = 0;
        v4i g2 = {0, 0, 0, 0};
        v4i g3 = {0, 0, 0, 0};
        v8i g4 = {0, 0, 0, 0, 0, 0, 0, 0};
        __builtin_amdgcn_tensor_load_to_lds(g0, g1, g2, g3, g4, 0);
        __builtin_amdgcn_s_wait_tensorcnt(0);
    }
#else
    if (tid < 4 * GBOX) yb[tid] = y[tid];
#endif
    for (int j = tid; j < M; j += 256) { scb[j] = ws[WS_CB + j]; sci[j] = ws[WS_CI + j]; }
    __syncthreads();

    float acc = 0.0f;
    for (int j = tid; j < M; j += 256) {
        unsigned bj = scb[j], ij = sci[j];
        int rank = 0;
        for (int m = 0; m < M; ++m) {
            unsigned bm = scb[m];
            rank += (int)((bm > bj) || (bm == bj && sci[m] < ij));
        }
        if (rank < KTOP) {
            size_t base = (size_t)ij * 6;
            float2 p01 = *(const float2*)(pred + base);
            float2 p23 = *(const float2*)(pred + base + 2);
            float4 a   = *(const float4*)(anch + (size_t)ij * 4);
            float xx = p01.x * a.z + a.x;
            float yy = p01.y * a.w + a.y;
            float w  = __expf(p23.x) * a.z;
            float h  = __expf(p23.y) * a.w;
            bool matched = false;
            #pragma unroll 8
            for (int g = 0; g < GBOX; ++g) {
                float y0 = yb[g*4+0], y1 = yb[g*4+1], y2 = yb[g*4+2], y3 = yb[g*4+3];
                float dw = fminf(xx + w, y0 + y2) - fmaxf(xx, y0);
                float dh = fminf(yy + h, y1 + y3) - fmaxf(yy, y1);
                float inter = dw * dh;
                float iou = inter / (w * h + y2 * y3 - inter);
                matched = matched || (iou > IOU_T);
            }
            float s = __uint_as_float(bj);
            acc += matched ? log1pf(-s) : logf(s);
        }
    }
    partial[tid] = acc;
    __syncthreads();

    // Reduce 256 partials with 4 chained V_WMMA_F32_16X16X4_F32 (B = ones).
    if (tid < 32) {
        v8f dacc = {0, 0, 0, 0, 0, 0, 0, 0};
        v2f ones; ones.x = 1.0f; ones.y = 1.0f;
        #pragma unroll
        for (int c = 0; c < 4; ++c) {
            v2f afrag;
            afrag.x = partial[c * 64 + tid * 2 + 0];
            afrag.y = partial[c * 64 + tid * 2 + 1];
            dacc = __builtin_amdgcn_wmma_f32_16x16x4_f32(
                false, afrag, false, ones, (short)0, dacc, false, false);
        }
        // Column N=0 of D: lane 0 holds rows 0..7, lane 16 holds rows 8..15.
        if (tid == 0 || tid == 16) {
            float t = dacc[0] + dacc[1] + dacc[2] + dacc[3]
                    + dacc[4] + dacc[5] + dacc[6] + dacc[7];
            colsum[tid >> 4] = t;
        }
    }
    __syncthreads();
    if (tid == 0) out[0] = colsum[0] + colsum[1];
}

extern "C" void kernel_launch(void* const* d_in, const int* in_sizes, int n_in,
                              void* d_out, int out_size, void* d_ws, size_t ws_size,
                              hipStream_t stream) {
    (void)in_sizes; (void)n_in; (void)out_size; (void)ws_size;
    const float* pred = (const float*)d_in[0];
    const float* anch = (const float*)d_in[1];
    const float* y    = (const float*)d_in[2];
    unsigned* ws = (unsigned*)d_ws;
    float* out = (float*)d_out;

    const int nzero = NBINS + 4;   // hist + sel + counter
    zero_ws<<<(nzero + 255) / 256, 256, 0, stream>>>(ws, nzero);
    score_hist<<<2048, 256, 0, stream>>>(pred, ws);
    select_threshold<<<1, 1024, 0, stream>>>(ws);
    compact_cand<<<2048, 256, 0, stream>>>(pred, ws);
    final_loss<<<1, 256, 0, stream>>>(pred, anch, y, ws, out);
}